// SupervisedL1RegLoss_46832323395803
// MI455X (gfx1250) — compile-verified
//
#include <hip/hip_runtime.h>

#define NB 4
#define NP 8192
#define KNN 32
#define KBQ 64
#define RR 0.5625f   // 0.75^2, exact in fp32

typedef __attribute__((ext_vector_type(2))) float v2f;
typedef __attribute__((ext_vector_type(8))) float v8f;
typedef __attribute__((ext_vector_type(4))) unsigned int v4u;
typedef __attribute__((ext_vector_type(8))) int v8i;
typedef __attribute__((ext_vector_type(4))) int v4i;

#define HAVE_TDM __has_builtin(__builtin_amdgcn_tensor_load_to_lds)

// ---------------- init / data-loss / finalize ----------------

__global__ void init_acc(float* acc) {
    if (threadIdx.x < 3) acc[threadIdx.x] = 0.0f;
}

__global__ void data_loss_kernel(const float* __restrict__ p,
                                 const float* __restrict__ g,
                                 float* __restrict__ acc, int n) {
    float s = 0.0f;
    for (int i = blockIdx.x * blockDim.x + threadIdx.x; i < n;
         i += gridDim.x * blockDim.x)
        s += fabsf(p[i] - g[i]);
    #pragma unroll
    for (int off = 16; off > 0; off >>= 1) s += __shfl_down(s, off, 32);
    if ((threadIdx.x & 31) == 0) atomicAdd(&acc[0], s);
}

__global__ void finalize_kernel(const float* __restrict__ acc, float* __restrict__ out) {
    if (threadIdx.x == 0) {
        const float inv_bn = 1.0f / (float)(NB * NP);
        const float data   = acc[0] * (inv_bn / 3.0f);
        const float knn    = acc[1] * inv_bn;
        const float bq     = acc[2] * inv_bn;
        out[0] = 0.75f * data + 0.125f * knn + 0.125f * bq;
    }
}

// ---------------- main: TDM LDS fill + WMMA Gram + KNN/ball-query ----------------
// grid: 512 blocks of 128 threads (4 waves). Each wave owns 16 rows of one batch.

__global__ __launch_bounds__(128) void flow_reg_kernel(
    const float* __restrict__ pc,    // (B,N,3)
    const float* __restrict__ flow,  // (B,N,3) pred_flow
    float* __restrict__ acc)
{
    __shared__ float spt[NP * 3 + 4];   // AoS xyz, +pad so paired loads stay in-bounds
    __shared__ float sq[NP];            // squared norms
    __shared__ float stage[4][16 * 16]; // per-wave 16x16 sqd staging

    const int b    = blockIdx.x >> 7;   // 128 blocks per batch
    const int t64  = blockIdx.x & 127;
    const int tid  = threadIdx.x;
    const int w    = tid >> 5;
    const int lane = tid & 31;

    const float* pcb = pc   + (size_t)b * NP * 3;
    const float* flb = flow + (size_t)b * NP * 3;

#if HAVE_TDM
    // Tensor Data Mover: one 1-D descriptor moves the whole 96KB point block
    // global -> LDS asynchronously (TENSORcnt-tracked).
    if (w == 0) {
        unsigned long long ga = (unsigned long long)(const void*)pcb;
        unsigned int lds_off  = (unsigned int)(size_t)(void*)spt;
        v4u g0;
        g0.x = 1u;                                   // count=1 (valid user D#)
        g0.y = lds_off;                              // lds_addr
        g0.z = (unsigned int)(ga & 0xffffffffull);   // global_addr[31:0]
        g0.w = (unsigned int)((ga >> 32) & 0x1ffffffull) | (2u << 30); // addr[56:32], type=2
        v8i g1;
        g1[0] = 0x00020000;          // workgroup_mask=0, data_size=2 (4 bytes)
        g1[1] = 0x60000000;          // tensor_dim0[15:0]=24576 in bits[63:48]
        g1[2] = 0x00010000;          // tensor_dim0[31:16]=0, tensor_dim1=1
        g1[3] = 0x60000000;          // tile_dim0=24576
        g1[4] = 0x00000001;          // tile_dim1=1, tile_dim2=0
        g1[5] = 0x00006000;          // tensor_dim0_stride=24576
        g1[6] = 0;
        g1[7] = 0;
        v4i gz4 = {0, 0, 0, 0};
        v8i gz8 = {0, 0, 0, 0, 0, 0, 0, 0};
        __builtin_amdgcn_tensor_load_to_lds(g0, g1, gz4, gz4, gz8, 0);
        __builtin_amdgcn_s_wait_tensorcnt(0);
    }
#else
    for (int e = tid; e < NP * 3; e += 128) spt[e] = pcb[e];
#endif
    __syncthreads();

    for (int p = tid; p < NP; p += 128) {
        float x = spt[3 * p + 0], y = spt[3 * p + 1], z = spt[3 * p + 2];
        sq[p] = x * x + y * y + z * z;
    }
    __syncthreads();

    const int  n0   = t64 * 64 + w * 16;
    const int  r    = lane & 15;
    const bool hi   = lane >= 16;
    const int  rowg = n0 + r;

    // WMMA A operand, 16x4 f32: lanes 0-15 (x,y) = K0,K1; lanes 16-31 (z,0) = K2,K3
    const int abase = 3 * rowg + (hi ? 2 : 0);
    float a0 = spt[abase], a1 = spt[abase + 1];
    v2f a;
    a.x = a0;
    a.y = hi ? 0.0f : a1;

    // |p|^2 for the 8 rows this lane holds in the C/D tile
    float rq[8];
    #pragma unroll
    for (int j = 0; j < 8; ++j) rq[j] = sq[n0 + j + (hi ? 8 : 0)];

    // flow of this lane's scan row
    const float frx = flb[3 * rowg + 0];
    const float fry = flb[3 * rowg + 1];
    const float frz = flb[3 * rowg + 2];

    // register-resident sorted top-32 (hi lanes: KNN for row r)
    float kd[KNN]; int ki[KNN];
    #pragma unroll
    for (int k = 0; k < KNN; ++k) { kd[k] = 3.0e38f; ki[k] = rowg; }
    float worst = 3.0e38f;

    // ball-query state (low lanes: BQ for row r, columns scanned index-ascending)
    int   bq_cnt = 0, bq_first = rowg;
    float bq_sum = 0.0f;

    float* st = &stage[w][0];

    for (int c0 = 0; c0 < NP; c0 += 16) {
        const int cl    = c0 + r;
        const int bbase = 3 * cl + (hi ? 2 : 0);
        float b0 = spt[bbase], b1 = spt[bbase + 1];
        v2f bb;
        bb.x = b0;
        bb.y = hi ? 0.0f : b1;

        v8f g = {};
        g = __builtin_amdgcn_wmma_f32_16x16x4_f32(false, a, false, bb,
                                                  (short)0, g, false, false);

        // sqd = |pr|^2 + |pc|^2 - 2*G, clamp at 0; stage to LDS as [row][col]
        const float cq = sq[cl];
        #pragma unroll
        for (int j = 0; j < 8; ++j) {
            float d = fmaf(-2.0f, g[j], rq[j] + cq);
            d = fmaxf(d, 0.0f);
            st[(j + (hi ? 8 : 0)) * 16 + r] = d;
        }
        __builtin_amdgcn_fence(__ATOMIC_RELEASE, "wavefront");
        __builtin_amdgcn_wave_barrier();

        if (!hi) {
            // ball query: first KBQ lowest-index in-radius columns
            for (int c = 0; c < 16; ++c) {
                float d = st[r * 16 + c];
                if (d < RR) {
                    int idx = c0 + c;
                    if (bq_cnt == 0) bq_first = idx;
                    if (bq_cnt < KBQ) {
                        float dx = frx - flb[3 * idx + 0];
                        float dy = fry - flb[3 * idx + 1];
                        float dz = frz - flb[3 * idx + 2];
                        bq_sum += fabsf(dx) + fabsf(dy) + fabsf(dz);
                    }
                    bq_cnt++;
                }
            }
        } else {
            // KNN: sorted-insert with early reject against current 32nd distance
            for (int c = 0; c < 16; ++c) {
                float d = st[r * 16 + c];
                if (d < worst) {
                    float dv = d; int iv = c0 + c;
                    #pragma unroll
                    for (int k = 0; k < KNN; ++k) {
                        bool  s  = dv < kd[k];
                        float td = kd[k]; int ti = ki[k];
                        kd[k] = s ? dv : td;  ki[k] = s ? iv : ti;
                        dv    = s ? td : dv;  iv    = s ? ti : iv;
                    }
                    worst = kd[KNN - 1];
                }
            }
        }
        __builtin_amdgcn_wave_barrier();
        __builtin_amdgcn_fence(__ATOMIC_ACQUIRE, "wavefront");
    }

    if (!hi) {
        float contrib = bq_sum;
        if (bq_cnt < KBQ) {   // pad with first in-radius index (reference semantics)
            float dx = frx - flb[3 * bq_first + 0];
            float dy = fry - flb[3 * bq_first + 1];
            float dz = frz - flb[3 * bq_first + 2];
            contrib += (float)(KBQ - bq_cnt) * (fabsf(dx) + fabsf(dy) + fabsf(dz));
        }
        atomicAdd(&acc[2], contrib * (1.0f / (float)KBQ));
    } else {
        float s = 0.0f;
        #pragma unroll
        for (int k = 0; k < KNN; ++k) {
            if (kd[k] <= RR) {  // dist > RADIUS -> replaced by self -> contributes 0
                float dx = frx - flb[3 * ki[k] + 0];
                float dy = fry - flb[3 * ki[k] + 1];
                float dz = frz - flb[3 * ki[k] + 2];
                s += fabsf(dx) + fabsf(dy) + fabsf(dz);
            }
        }
        atomicAdd(&acc[1], s * (1.0f / (float)KNN));
    }
}

// ---------------- launch ----------------

extern "C" void kernel_launch(void* const* d_in, const int* in_sizes, int n_in,
                              void* d_out, int out_size, void* d_ws, size_t ws_size,
                              hipStream_t stream) {
    const float* pc_source = (const float*)d_in[0];
    // d_in[1] = pc_target (unused by reference)
    const float* pred_flow = (const float*)d_in[2];
    const float* gt_flow   = (const float*)d_in[3];
    float* out = (float*)d_out;
    float* acc = (float*)d_ws;   // [0]=data sum, [1]=knn sum, [2]=bq sum

    init_acc<<<1, 32, 0, stream>>>(acc);
    data_loss_kernel<<<192, 256, 0, stream>>>(pred_flow, gt_flow, acc, NB * NP * 3);
    flow_reg_kernel<<<NB * (NP / 64), 128, 0, stream>>>(pc_source, pred_flow, acc);
    finalize_kernel<<<1, 32, 0, stream>>>(acc, out);
}